// ONINorm_colum_12421045420977
// MI455X (gfx1250) — compile-verified
//
#include <hip/hip_runtime.h>

// ---------------------------------------------------------------------------
// ONI norm (Newton-Schulz T=2) for weight (8192,1024), fp32, 32 groups.
// Per group g: Z (256x1024), Zc = Z - rowmean(Z)
//   S  = Zc^T Zc + eps*I                  (1024x1024, K=256)   [WMMA f32]
//   sumSq[g] = ||S||_F^2                  (fused in S epilogue, atomics)
//   B1 = 1.5 I - 0.5 S                    (elementwise; exact since B0 = I)
//   M  = B1 @ B1                          [WMMA f32]
//   B3 = M  @ B1                          [WMMA f32]
//   B2 = 1.5 B1 - 0.5 (B3 @ S)            [WMMA f32, fused epilogue]
//   W  = (Zc @ B2) * rsqrt(sqrt(sumSq))   [WMMA f32, fused scale]
// ---------------------------------------------------------------------------

typedef __attribute__((ext_vector_type(2))) float v2f;
typedef __attribute__((ext_vector_type(8))) float v8f;

#define TILE_M 128
#define TILE_N 128
#define KCHUNK 16  // 4 quads of K=4
#define ONI_EPS 1e-5f

// ---------------------------------------------------------------------------
// Batched fp32 WMMA GEMM:  C = alpha * op(A) @ B  [+ beta * E] [+ eps diag]
//                          [* rsqrt(sqrt(sumSqIn[g]))] ; optional Frobenius
//                          sum-of-squares accumulation into sumSqOut[g].
//
// Block: 256 threads (8 waves), tile 128x128; wave tile 64x32 (4x2 WMMA).
// LDS layout stores each K-quad contiguously per row: element (k, m) lives at
// As[k>>2][m][k&3], so a WMMA f32 operand pair (k, k+1) for a lane is ONE
// 8-byte-aligned ds_load_b64 straight into an even VGPR pair (no repacking).
// Lanes 0-15 (kHalf=0) hit banks {4m,4m+1}, lanes 16-31 (kHalf=2) hit
// {4m+2,4m+3}: all 64 banks once -> conflict-free fragment loads.
// All global->LDS stores are b128 of a full quad.
// ---------------------------------------------------------------------------
template <bool TRANSA>
__global__ __launch_bounds__(256) void gemm_wmma_f32(
    const float* __restrict__ A, int lda, size_t strideA,
    const float* __restrict__ B, int ldb, size_t strideB,
    const float* __restrict__ E, size_t strideE,
    float* __restrict__ C, int ldc, size_t strideC,
    int K, float alpha, float beta,
    const float* __restrict__ sumSqIn,  // per-group scale source (may be null)
    float* __restrict__ sumSqOut)       // per-group fro accum (may be null)
{
  __shared__ float As[4][TILE_M + 1][4];  // [k>>2][m][k&3], +1 row pad
  __shared__ float Bs[4][TILE_N + 1][4];  // [k>>2][n][k&3]
  __shared__ float red[256];

  const int g = blockIdx.z;
  A += (size_t)g * strideA;
  B += (size_t)g * strideB;
  C += (size_t)g * strideC;
  if (E) E += (size_t)g * strideE;

  const int tid  = threadIdx.x;
  const int row0 = blockIdx.y * TILE_M;
  const int col0 = blockIdx.x * TILE_N;

  const int wave  = tid >> 5;
  const int lane  = tid & 31;
  const int mBase = (wave >> 2) * 64;  // 2 waves along M
  const int nBase = (wave & 3) * 32;   // 4 waves along N
  const int l15   = lane & 15;
  const int hi    = lane >> 4;
  const int kHalf = hi << 1;           // 0 or 2 (f32 A/B K-pair layout)

  v8f acc[4][2] = {};

  for (int k0 = 0; k0 < K; k0 += KCHUNK) {
    // ---- global -> LDS
    if (TRANSA) {
      // A stored K x M (Zc as r x d); 4 coalesced column reads -> b128 store
      #pragma unroll
      for (int r = 0; r < 2; ++r) {
        const int idx = tid + r * 256;  // 0..511
        const int m   = idx & 127;
        const int kq  = idx >> 7;       // 0..3
        const float* p = A + (size_t)(k0 + kq * 4) * lda + (row0 + m);
        float4 v;
        v.x = p[0];
        v.y = p[(size_t)lda];
        v.z = p[(size_t)lda * 2];
        v.w = p[(size_t)lda * 3];
        *(float4*)(&As[kq][m][0]) = v;
      }
    } else {
      // A stored M x K; one float4 = one K-quad of a row -> b128 store
      #pragma unroll
      for (int r = 0; r < 2; ++r) {
        const int idx = tid + r * 256;
        const int mr  = idx >> 2;            // 0..127
        const int qq  = idx & 3;             // 0..3
        const float4 v =
            *(const float4*)(A + (size_t)(row0 + mr) * lda + (k0 + qq * 4));
        *(float4*)(&As[qq][mr][0]) = v;
      }
    }
    #pragma unroll
    for (int r = 0; r < 2; ++r) {
      const int idx = tid + r * 256;
      const int n   = idx & 127;
      const int kq  = idx >> 7;
      const float* p = B + (size_t)(k0 + kq * 4) * ldb + (col0 + n);
      float4 v;
      v.x = p[0];
      v.y = p[(size_t)ldb];
      v.z = p[(size_t)ldb * 2];
      v.w = p[(size_t)ldb * 3];
      *(float4*)(&Bs[kq][n][0]) = v;
    }
    __syncthreads();

    // ---- compute: 4 k-steps (K=4 each) of V_WMMA_F32_16X16X4_F32
    #pragma unroll
    for (int q = 0; q < 4; ++q) {
      v2f a[4], b[2];
      #pragma unroll
      for (int mi = 0; mi < 4; ++mi)
        a[mi] = *(const v2f*)(&As[q][mBase + mi * 16 + l15][kHalf]);
      #pragma unroll
      for (int ni = 0; ni < 2; ++ni)
        b[ni] = *(const v2f*)(&Bs[q][nBase + ni * 16 + l15][kHalf]);
      #pragma unroll
      for (int mi = 0; mi < 4; ++mi)
        #pragma unroll
        for (int ni = 0; ni < 2; ++ni)
          acc[mi][ni] = __builtin_amdgcn_wmma_f32_16x16x4_f32(
              false, a[mi], false, b[ni], (short)0, acc[mi][ni], false, false);
    }
    __syncthreads();
  }

  // ---- epilogue
  float scale = 1.0f;
  if (sumSqIn) scale = rsqrtf(sqrtf(sumSqIn[g]));  // 1/sqrt(||S||_F)

  float localSq = 0.0f;
  #pragma unroll
  for (int mi = 0; mi < 4; ++mi) {
    #pragma unroll
    for (int ni = 0; ni < 2; ++ni) {
      #pragma unroll
      for (int v = 0; v < 8; ++v) {
        // C/D layout: VGPR v -> M = v + 8*hi ; N = lane&15
        const int row = row0 + mBase + mi * 16 + v + 8 * hi;
        const int col = col0 + nBase + ni * 16 + l15;
        float val = alpha * acc[mi][ni][v];
        if (E) val += beta * E[(size_t)row * ldc + col];
        if (sumSqOut) {
          if (row == col) val += ONI_EPS;
          localSq += val * val;
        }
        val *= scale;
        C[(size_t)row * ldc + col] = val;
      }
    }
  }

  if (sumSqOut) {
    red[tid] = localSq;
    __syncthreads();
    for (int off = 128; off > 0; off >>= 1) {
      if (tid < off) red[tid] += red[tid + off];
      __syncthreads();
    }
    if (tid == 0) atomicAdd(&sumSqOut[g], red[0]);
  }
}

// ---------------------------------------------------------------------------
__global__ __launch_bounds__(256) void center_rows_kernel(
    const float* __restrict__ W, float* __restrict__ Zc) {
  __shared__ float red[256];
  const int row = blockIdx.x;
  const int tid = threadIdx.x;
  const float* src = W + (size_t)row * 1024;
  float* dst = Zc + (size_t)row * 1024;

  float s = 0.0f;
  #pragma unroll
  for (int i = 0; i < 4; ++i) s += src[tid + i * 256];
  red[tid] = s;
  __syncthreads();
  for (int off = 128; off > 0; off >>= 1) {
    if (tid < off) red[tid] += red[tid + off];
    __syncthreads();
  }
  const float mean = red[0] * (1.0f / 1024.0f);
  #pragma unroll
  for (int i = 0; i < 4; ++i)
    dst[tid + i * 256] = src[tid + i * 256] - mean;
}

__global__ void b1_kernel(const float* __restrict__ S, float* __restrict__ B1,
                          long long n) {
  long long i = (long long)blockIdx.x * blockDim.x + threadIdx.x;
  const long long stride = (long long)gridDim.x * blockDim.x;
  for (; i < n; i += stride) {
    const int li = (int)(i & (1024 * 1024 - 1));
    float v = -0.5f * S[i];
    if ((li >> 10) == (li & 1023)) v += 1.5f;
    B1[i] = v;
  }
}

__global__ void zero32_kernel(float* p) {
  if (threadIdx.x < 32) p[threadIdx.x] = 0.0f;
}

// ---------------------------------------------------------------------------
extern "C" void kernel_launch(void* const* d_in, const int* in_sizes, int n_in,
                              void* d_out, int out_size, void* d_ws,
                              size_t ws_size, hipStream_t stream) {
  (void)in_sizes; (void)n_in; (void)out_size;
  const float* Win = (const float*)d_in[0];
  float* Wout = (float*)d_out;

  const int G = 32, R = 256, D = 1024;
  const size_t DD = (size_t)D * D;                 // elems per group matrix
  const size_t matBytes = DD * sizeof(float);      // 4 MB
  const size_t zcElems = (size_t)8192 * 1024;
  const size_t zcBytes = zcElems * sizeof(float);  // 32 MB

  char* ws = (char*)d_ws;
  float* sumSq = (float*)ws;                       // 32 floats (+pad to 256B)
  float* Zc = (float*)(ws + 256);
  char* pool = ws + 256 + zcBytes;

  size_t remain = (ws_size > 256 + zcBytes) ? (ws_size - 256 - zcBytes) : 0;
  int gc = (int)(remain / (4 * matBytes));  // groups per chunk
  if (gc < 1) gc = 1;
  if (gc > G) gc = G;

  zero32_kernel<<<1, 32, 0, stream>>>(sumSq);
  center_rows_kernel<<<8192, 256, 0, stream>>>(Win, Zc);

  for (int g0 = 0; g0 < G; g0 += gc) {
    const int gn = (G - g0 < gc) ? (G - g0) : gc;
    float* S  = (float*)pool;
    float* B1 = S  + (size_t)gc * DD;
    float* M1 = B1 + (size_t)gc * DD;
    float* B3 = M1 + (size_t)gc * DD;
    float* B2 = M1;  // reuse: M1 dead after B3 = M1 @ B1

    const float* ZcG = Zc + (size_t)g0 * R * D;
    const size_t sZc = (size_t)R * D;

    // S = Zc^T Zc + eps*I ; sumSq[g] += ||S||_F^2
    gemm_wmma_f32<true><<<dim3(D / TILE_N, D / TILE_M, gn), 256, 0, stream>>>(
        ZcG, D, sZc, ZcG, D, sZc, nullptr, 0, S, D, DD, R, 1.0f, 0.0f,
        nullptr, sumSq + g0);

    // B1 = 1.5 I - 0.5 S
    b1_kernel<<<4096, 256, 0, stream>>>(S, B1, (long long)gn * DD);

    // M1 = B1 @ B1
    gemm_wmma_f32<false><<<dim3(D / TILE_N, D / TILE_M, gn), 256, 0, stream>>>(
        B1, D, DD, B1, D, DD, nullptr, 0, M1, D, DD, D, 1.0f, 0.0f,
        nullptr, nullptr);

    // B3 = M1 @ B1
    gemm_wmma_f32<false><<<dim3(D / TILE_N, D / TILE_M, gn), 256, 0, stream>>>(
        M1, D, DD, B1, D, DD, nullptr, 0, B3, D, DD, D, 1.0f, 0.0f,
        nullptr, nullptr);

    // B2 = 1.5 B1 - 0.5 (B3 @ S)
    gemm_wmma_f32<false><<<dim3(D / TILE_N, D / TILE_M, gn), 256, 0, stream>>>(
        B3, D, DD, S, D, DD, B1, DD, B2, D, DD, D, -0.5f, 1.5f,
        nullptr, nullptr);

    // W = (Zc @ B2) * rsqrt(sqrt(sumSq[g]))
    gemm_wmma_f32<false><<<dim3(D / TILE_N, R / TILE_M, gn), 256, 0, stream>>>(
        ZcG, D, sZc, B2, D, DD, nullptr, 0, Wout + (size_t)g0 * R * D, D, sZc,
        D, 1.0f, 0.0f, sumSq + g0, nullptr);
  }
}